// M_39745627357299
// MI455X (gfx1250) — compile-verified
//
#include <hip/hip_runtime.h>

typedef float v2f __attribute__((ext_vector_type(2)));
typedef float v8f __attribute__((ext_vector_type(8)));

#define S_LEN   2048
#define HEADS   12
#define HD      64
#define C3      2304            // 3*768 input row stride
#define QT      16              // q rows per wave
#define NT      128             // S/16 tiles
#define TSTR    20              // LDS transpose-tile stride (bank-conflict-free)
#define NEG_MAX (-3.402823466e38f)

// ---- cross-lane row reductions: C-layout puts one row across 16 lanes ----
__device__ __forceinline__ float rmax16(float v) {
    v = fmaxf(v, __shfl_xor(v, 1, 32));
    v = fmaxf(v, __shfl_xor(v, 2, 32));
    v = fmaxf(v, __shfl_xor(v, 4, 32));
    v = fmaxf(v, __shfl_xor(v, 8, 32));
    return v;
}
__device__ __forceinline__ float rsum16(float v) {
    v += __shfl_xor(v, 1, 32);
    v += __shfl_xor(v, 2, 32);
    v += __shfl_xor(v, 4, 32);
    v += __shfl_xor(v, 8, 32);
    return v;
}

// ================= Kernel A: online softmax stats (rowmax, 1/rowsum) =========
// 4 waves / block, one 16-row q-tile per wave, zero LDS -> high occupancy.
extern "C" __global__ __launch_bounds__(128)
void attn_stats_kernel(const float* __restrict__ x,
                       float* __restrict__ smax, float* __restrict__ sinvp)
{
    const int lane = threadIdx.x & 31;
    const int wid  = threadIdx.x >> 5;
    const int half = lane >> 4;
    const int n16  = lane & 15;

    const int g  = blockIdx.x * 4 + wid;       // 4 consecutive q-tiles share (b,h)
    const int qt = g & (NT - 1);
    const int h  = (g >> 7) % HEADS;
    const int b  = g / (NT * HEADS);
    const int q0 = qt * QT;

    // Q tile in A-operand layout: lane row m=n16, k = 4c + 2*half + v
    const float* qptr = x + ((size_t)b * S_LEN + q0 + n16) * C3 + h * HD + 2 * half;
    v2f qa[16];
#pragma unroll
    for (int c = 0; c < 16; ++c) qa[c] = *(const v2f*)(qptr + 4 * c);

    const float* kbase = x + (size_t)b * S_LEN * C3 + 768 + h * HD + 2 * half;

    float runmax[8], runsum[8];
#pragma unroll
    for (int i = 0; i < 8; ++i) { runmax[i] = NEG_MAX; runsum[i] = 0.0f; }

    for (int kt = 0; kt <= qt; ++kt) {
        const int k0 = kt * QT;
        const float* kptr = kbase + (size_t)(k0 + n16) * C3;
        if (kt < qt) __builtin_prefetch(kptr + QT * C3, 0, 0);

        v2f kb[16];
#pragma unroll
        for (int c = 0; c < 16; ++c) kb[c] = *(const v2f*)(kptr + 4 * c);

        v8f acc = {};
#pragma unroll
        for (int c = 0; c < 16; ++c)
            acc = __builtin_amdgcn_wmma_f32_16x16x4_f32(
                false, qa[c], false, kb[c], (short)0, acc, false, false);

#pragma unroll
        for (int i = 0; i < 8; ++i) {
            const int r = i + 8 * half;
            float s = acc[i] * 0.125f;
            s = (k0 + n16 > q0 + r) ? NEG_MAX : s;        // causal mask
            const float tmax = rmax16(s);
            const float nmax = fmaxf(runmax[i], tmax);
            const float p    = __expf(s - nmax);
            const float tsum = rsum16(p);
            runsum[i] = runsum[i] * __expf(runmax[i] - nmax) + tsum;
            runmax[i] = nmax;
        }
    }

    if (n16 == 0) {                                        // lanes 0 and 16
        const size_t base = (size_t)(b * HEADS + h) * S_LEN + q0 + 8 * half;
#pragma unroll
        for (int i = 0; i < 8; ++i) {
            smax [base + i] = runmax[i];
            sinvp[base + i] = 1.0f / runsum[i];
        }
    }
}

// ===== Kernel B: recompute QK^T, write normalized weights, PV, output ========
extern "C" __global__ __launch_bounds__(128)
void attn_main_kernel(const float* __restrict__ x,
                      const float* __restrict__ smax, const float* __restrict__ sinvp,
                      float* __restrict__ outO, float* __restrict__ outW)
{
    __shared__ float ptile[4][QT * TSTR];    // per-wave C->A transpose tile (1.25KB)

    const int lane = threadIdx.x & 31;
    const int wid  = threadIdx.x >> 5;
    const int half = lane >> 4;
    const int n16  = lane & 15;

    const int g  = blockIdx.x * 4 + wid;
    const int qt = g & (NT - 1);
    const int h  = (g >> 7) % HEADS;
    const int b  = g / (NT * HEADS);
    const int q0 = qt * QT;

    float* tp = ptile[wid];

    const float* qptr = x + ((size_t)b * S_LEN + q0 + n16) * C3 + h * HD + 2 * half;
    v2f qa[16];
#pragma unroll
    for (int c = 0; c < 16; ++c) qa[c] = *(const v2f*)(qptr + 4 * c);

    const float* kbase = x + (size_t)b * S_LEN * C3 + 768  + h * HD + 2 * half;
    const float* vbase = x + (size_t)b * S_LEN * C3 + 1536 + h * HD + n16;

    // per-lane row stats for rows r = i + 8*half (broadcast loads)
    const size_t sbase = (size_t)(b * HEADS + h) * S_LEN + q0 + 8 * half;
    float mx[8], inv[8];
#pragma unroll
    for (int i = 0; i < 8; ++i) { mx[i] = smax[sbase + i]; inv[i] = sinvp[sbase + i]; }

    float* wbase = outW + ((size_t)(b * HEADS + h) * S_LEN + q0) * S_LEN;

    v8f o0 = {}, o1 = {}, o2 = {}, o3 = {};

    for (int kt = 0; kt < NT; ++kt) {
        const int k0 = kt * QT;
        if (kt <= qt) {                                    // wave-uniform branch
            const float* kptr = kbase + (size_t)(k0 + n16) * C3;
            v2f kb[16];
#pragma unroll
            for (int c = 0; c < 16; ++c) kb[c] = *(const v2f*)(kptr + 4 * c);

            v8f acc = {};
#pragma unroll
            for (int c = 0; c < 16; ++c)
                acc = __builtin_amdgcn_wmma_f32_16x16x4_f32(
                    false, qa[c], false, kb[c], (short)0, acc, false, false);

            // p = exp(s - rowmax); weights out; stage C->A transpose in LDS
#pragma unroll
            for (int i = 0; i < 8; ++i) {
                const int r = i + 8 * half;
                float s = acc[i] * 0.125f;
                s = (k0 + n16 > q0 + r) ? NEG_MAX : s;     // masked -> p == 0
                const float p = __expf(s - mx[i]);
                wbase[(size_t)r * S_LEN + k0 + n16] = p * inv[i];
                tp[r * TSTR + n16] = p;
            }
            // same-wave LDS ops are in-order (DScnt); fence for cross-lane read
            asm volatile("s_wait_dscnt 0" ::: "memory");

            // PV: A = P from LDS (lane row m=n16), B = V rows from global
            const float* arow = tp + n16 * TSTR + 2 * half;
#pragma unroll
            for (int c = 0; c < 4; ++c) {
                const v2f a = *(const v2f*)(arow + 4 * c);
                const size_t r0 = (size_t)(k0 + 4 * c + 2 * half) * C3;
                const v2f b0 = { vbase[r0 +  0], vbase[r0 + C3 +  0] };
                const v2f b1 = { vbase[r0 + 16], vbase[r0 + C3 + 16] };
                const v2f b2 = { vbase[r0 + 32], vbase[r0 + C3 + 32] };
                const v2f b3 = { vbase[r0 + 48], vbase[r0 + C3 + 48] };
                o0 = __builtin_amdgcn_wmma_f32_16x16x4_f32(false, a, false, b0, (short)0, o0, false, false);
                o1 = __builtin_amdgcn_wmma_f32_16x16x4_f32(false, a, false, b1, (short)0, o1, false, false);
                o2 = __builtin_amdgcn_wmma_f32_16x16x4_f32(false, a, false, b2, (short)0, o2, false, false);
                o3 = __builtin_amdgcn_wmma_f32_16x16x4_f32(false, a, false, b3, (short)0, o3, false, false);
            }
        } else {
            // strictly-above-diagonal tile: weights are exactly zero
#pragma unroll
            for (int i = 0; i < 8; ++i)
                wbase[(size_t)(i + 8 * half) * S_LEN + k0 + n16] = 0.0f;
        }
    }

    // normalize and write attn_output
    float* obase = outO + ((size_t)(b * HEADS + h) * S_LEN + q0) * HD + n16;
#pragma unroll
    for (int i = 0; i < 8; ++i) {
        const int r = i + 8 * half;
        float* op = obase + (size_t)r * HD;
        op[ 0] = o0[i] * inv[i];
        op[16] = o1[i] * inv[i];
        op[32] = o2[i] * inv[i];
        op[48] = o3[i] * inv[i];
    }
}

extern "C" void kernel_launch(void* const* d_in, const int* in_sizes, int n_in,
                              void* d_out, int out_size, void* d_ws, size_t ws_size,
                              hipStream_t stream) {
    (void)in_sizes; (void)n_in; (void)ws_size; (void)out_size;
    const float* x = (const float*)d_in[0];          // [4,2048,2304] f32
    // d_in[1] = causal mask; causality applied analytically.
    float* outO = (float*)d_out;                                  // attn_output
    float* outW = outO + (size_t)4 * HEADS * S_LEN * HD;          // attn_weights

    float* smax = (float*)d_ws;                                   // [B*H*S]
    float* sinv = smax + (size_t)4 * HEADS * S_LEN;               // [B*H*S]

    const int grid = (4 * HEADS * NT) / 4;           // 1536 blocks x 4 waves
    attn_stats_kernel<<<grid, 128, 0, stream>>>(x, smax, sinv);
    attn_main_kernel <<<grid, 128, 0, stream>>>(x, smax, sinv, outO, outW);
}